// NTMMemory_6657199309158
// MI455X (gfx1250) — compile-verified
//
#include <hip/hip_runtime.h>
#include <math.h>

#define EPSF 1e-16f
#define BATCH 64
#define NROWS 16384
#define MDIM 64

typedef float v2f __attribute__((ext_vector_type(2)));
typedef float v8f __attribute__((ext_vector_type(8)));

// ---------------------------------------------------------------------------
// K1: content addressing.  One wave32 per 16-row tile of one batch.
// D = A x B with A = k replicated over the 16 M-rows, B = memory tile^T,
// so acc[0] in every lane == dot(mem_row(lane&15), k).  Row norms via VALU
// on the same registers + one half-wave shuffle.
// ---------------------------------------------------------------------------
__global__ __launch_bounds__(256) void ntm_content_kernel(
    const float* __restrict__ memory, const float* __restrict__ k,
    const float* __restrict__ beta, float* __restrict__ a_out) {
  const int wave = threadIdx.x >> 5;
  const int lane = threadIdx.x & 31;
  const int tile = blockIdx.x * (blockDim.x >> 5) + wave;
  const int b  = tile >> 10;           // NROWS/16 == 1024 tiles per batch
  const int nb = (tile & 1023) << 4;   // first row of this 16-row tile
  const int half = lane >> 4;          // 0: K={0,1}, 1: K={2,3}
  const int r = lane & 15;

  const float* mrow = memory + ((size_t)b * NROWS + nb + r) * MDIM;
  const float* kb   = k + b * MDIM;

  v8f acc = {};
  float sq = 0.f;   // sum over this lane's columns of mem_e^2
  float kn = 0.f;   // sum over this lane's columns of k_e^2
#pragma unroll
  for (int mb = 0; mb < MDIM; mb += 4) {
    const int c0 = mb + (half << 1);
    float2 mv = *(const float2*)(mrow + c0);   // 8B per lane, rows contiguous
    float b0 = mv.x + EPSF;
    float b1 = mv.y + EPSF;
    float a0 = kb[c0]     + EPSF;
    float a1 = kb[c0 + 1] + EPSF;
    v2f A  = {a0, a1};                          // A[m, K] = k_e[K] for all m
    v2f Bm = {b0, b1};                          // B[K, n] = mem_e[n, K]
    acc = __builtin_amdgcn_wmma_f32_16x16x4_f32(
        false, A, false, Bm, (short)0, acc, false, false);
    sq = fmaf(b0, b0, fmaf(b1, b1, sq));
    kn = fmaf(a0, a0, fmaf(a1, a1, kn));
  }
  // lane L covers columns {4j,4j+1}, lane L+16 covers {4j+2,4j+3}: combine.
  sq += __shfl_xor(sq, 16, 32);
  kn += __shfl_xor(kn, 16, 32);

  float num   = acc[0];                       // broadcast into both halves
  float denom = sqrtf(sq) * sqrtf(kn);
  float cosv  = num / fmaxf(denom, 1e-8f);
  if (lane < 16)
    a_out[(size_t)b * NROWS + nb + r] = beta[b] * cosv;
}

// ---------------------------------------------------------------------------
// Block reductions (1024 threads == 32 waves of wave32)
// ---------------------------------------------------------------------------
__device__ inline float waveSum(float v) {
#pragma unroll
  for (int o = 16; o > 0; o >>= 1) v += __shfl_down(v, o, 32);
  return v;
}
__device__ inline float waveMax(float v) {
#pragma unroll
  for (int o = 16; o > 0; o >>= 1) v = fmaxf(v, __shfl_down(v, o, 32));
  return v;
}
__device__ inline float blockSum(float v, float* red) {
  const int lane = threadIdx.x & 31, wid = threadIdx.x >> 5;
  v = waveSum(v);
  if (lane == 0) red[wid] = v;
  __syncthreads();
  if (wid == 0) {
    float x = waveSum(red[lane]);
    if (lane == 0) red[0] = x;
  }
  __syncthreads();
  float r = red[0];
  __syncthreads();
  return r;
}
__device__ inline float blockMax(float v, float* red) {
  const int lane = threadIdx.x & 31, wid = threadIdx.x >> 5;
  v = waveMax(v);
  if (lane == 0) red[wid] = v;
  __syncthreads();
  if (wid == 0) {
    float x = waveMax(red[lane]);
    if (lane == 0) red[0] = x;
  }
  __syncthreads();
  float r = red[0];
  __syncthreads();
  return r;
}

// ---------------------------------------------------------------------------
// K2: softmax -> gate -> circular 3-tap shift (via LDS) -> sharpen -> norm.
// One block per batch; wg row (64 KB) staged in dynamic LDS.
// ---------------------------------------------------------------------------
__global__ __launch_bounds__(1024) void ntm_post_kernel(
    const float* __restrict__ a_in, const float* __restrict__ gv,
    const float* __restrict__ s, const float* __restrict__ gamma,
    const float* __restrict__ w_prev, float* __restrict__ out) {
  const int b   = blockIdx.x;
  const int tid = threadIdx.x;
  const int T   = 1024;
  const int NPT = NROWS / T;  // 16 elements per thread

  extern __shared__ float wgS[];          // NROWS floats
  __shared__ float red[32];

  const float* aRow = a_in + (size_t)b * NROWS;
  const float* wpRow = w_prev + (size_t)b * NROWS;
  const float g  = gv[b];
  const float s0 = s[b * 3 + 0], s1 = s[b * 3 + 1], s2 = s[b * 3 + 2];
  const float gam = gamma[b];

  float av[NPT];
  float lmax = -INFINITY;
#pragma unroll
  for (int j = 0; j < NPT; ++j) {
    const int i = tid + j * T;            // coalesced
    av[j] = aRow[i];
    lmax = fmaxf(lmax, av[j]);
  }
  const float bmax = blockMax(lmax, red);

  float lsum = 0.f;
#pragma unroll
  for (int j = 0; j < NPT; ++j) {
    av[j] = __expf(av[j] - bmax);
    lsum += av[j];
  }
  const float inv = 1.f / blockSum(lsum, red);

#pragma unroll
  for (int j = 0; j < NPT; ++j) {
    const int i = tid + j * T;
    const float wc = av[j] * inv;
    wgS[i] = g * wc + (1.f - g) * wpRow[i];
  }
  __syncthreads();

  float lsum2 = 0.f;
#pragma unroll
  for (int j = 0; j < NPT; ++j) {
    const int i = tid + j * T;
    const float left  = wgS[(i == 0) ? (NROWS - 1) : (i - 1)];
    const float mid   = wgS[i];
    const float right = wgS[(i == NROWS - 1) ? 0 : (i + 1)];
    const float wh = left * s0 + mid * s1 + right * s2;
    const float w  = powf(wh, gam);       // wh >= 0
    av[j] = w;
    lsum2 += w;
  }
  const float invt = 1.f / (blockSum(lsum2, red) + EPSF);

#pragma unroll
  for (int j = 0; j < NPT; ++j) {
    const int i = tid + j * T;
    out[(size_t)b * NROWS + i] = av[j] * invt;
  }
}

// ---------------------------------------------------------------------------
extern "C" void kernel_launch(void* const* d_in, const int* in_sizes, int n_in,
                              void* d_out, int out_size, void* d_ws, size_t ws_size,
                              hipStream_t stream) {
  const float* memory = (const float*)d_in[0];  // (B, N, M)
  const float* k      = (const float*)d_in[1];  // (B, M)
  const float* beta   = (const float*)d_in[2];  // (B, 1)
  const float* g      = (const float*)d_in[3];  // (B, 1)
  const float* s      = (const float*)d_in[4];  // (B, 3)
  const float* gamma  = (const float*)d_in[5];  // (B, 1)
  const float* w_prev = (const float*)d_in[6];  // (B, N)
  float* out  = (float*)d_out;                  // (B, N)
  float* a_ws = (float*)d_ws;                   // B*N floats (4 MB)

  const int tilesPerBatch = NROWS / 16;                  // 1024
  const int totalTiles    = BATCH * tilesPerBatch;       // 65536
  const int wavesPerBlock = 8;                           // 256 threads
  dim3 grid1(totalTiles / wavesPerBlock);                // 8192 blocks

  ntm_content_kernel<<<grid1, 256, 0, stream>>>(memory, k, beta, a_ws);

  const size_t shmem = (size_t)NROWS * sizeof(float);    // 64 KB
  ntm_post_kernel<<<BATCH, 1024, shmem, stream>>>(a_ws, g, s, gamma, w_prev, out);
}